// InertiaRotation_62861141344414
// MI455X (gfx1250) — compile-verified
//
#include <hip/hip_runtime.h>

typedef float v2f __attribute__((ext_vector_type(2)));
typedef float v8f __attribute__((ext_vector_type(8)));

#define BB 32     // batches per block
#define XS 100    // padded floats per batch in s_xyz (96 + 4 pad: bank spread + 16B align)

__device__ inline void eigvec3(float a00, float a01, float a02,
                               float a11, float a12, float a22,
                               float lam, float& vx, float& vy, float& vz) {
    float m00 = a00 - lam, m11 = a11 - lam, m22 = a22 - lam;
    // rows of (A - lam*I): r0=(m00,a01,a02) r1=(a01,m11,a12) r2=(a02,a12,m22)
    float c0x = a01*a12 - a02*m11, c0y = a02*a01 - m00*a12, c0z = m00*m11 - a01*a01; // r0 x r1
    float c1x = a01*m22 - a02*a12, c1y = a02*a02 - m00*m22, c1z = m00*a12 - a01*a02; // r0 x r2
    float c2x = m11*m22 - a12*a12, c2y = a12*a02 - a01*m22, c2z = a01*a12 - m11*a02; // r1 x r2
    float n0 = c0x*c0x + c0y*c0y + c0z*c0z;
    float n1 = c1x*c1x + c1y*c1y + c1z*c1z;
    float n2 = c2x*c2x + c2y*c2y + c2z*c2z;
    float bx = c0x, by = c0y, bz = c0z, bn = n0;
    if (n1 > bn) { bx = c1x; by = c1y; bz = c1z; bn = n1; }
    if (n2 > bn) { bx = c2x; by = c2y; bz = c2z; bn = n2; }
    float inv = rsqrtf(bn + 1e-30f);
    vx = bx * inv; vy = by * inv; vz = bz * inv;
}

__global__ __launch_bounds__(256)
void inertia_rot_kernel(const float* __restrict__ xyz,
                        const float* __restrict__ mass,
                        float* __restrict__ out) {
    __shared__ float s_xyz[BB * XS];      // 3200 f  (12.8 KB)
    __shared__ float s_m[128];            // normalized masses (R=4, N=32)
    __shared__ float s_rot[BB * 4 * 9];   // 1152 f  (4.6 KB)
    __shared__ float s_out[BB * 384];     // 12288 f (48 KB) -> coalesced store stage

    const int tid = threadIdx.x;
    const size_t b0 = (size_t)blockIdx.x * BB;

    if (tid < 128) s_m[tid] = mass[tid];

    // ---- phase 1: coalesced b128 load of xyz tile into padded LDS ----
    const float4* gx = (const float4*)(xyz + b0 * 96);
    #pragma unroll
    for (int it = 0; it < 3; ++it) {
        int i = tid + it * 256;              // 0..767 float4s
        float4 v = gx[i];
        int bl = i / 24;
        int w  = i - bl * 24;
        *(float4*)&s_xyz[bl * XS + w * 4] = v;
    }
    __syncthreads();

    // mass normalization: m[r] /= sum_n m[r][n]
    if (tid < 4) {
        float s = 0.f;
        #pragma unroll
        for (int n = 0; n < 32; ++n) s += s_m[tid * 32 + n];
        float inv = 1.0f / s;
        #pragma unroll
        for (int n = 0; n < 32; ++n) s_m[tid * 32 + n] *= inv;
    }
    __syncthreads();

    // ---- phase 2: inertia tensor + symmetric 3x3 eigensolve, thread per (b,r) ----
    if (tid < BB * 4) {
        int bl = tid >> 2, r = tid & 3;
        const float* xp = &s_xyz[bl * XS];
        const float* mp = &s_m[r * 32];
        float sxx = 0, syy = 0, szz = 0, sxy = 0, sxz = 0, syz = 0;
        #pragma unroll
        for (int n = 0; n < 32; ++n) {
            float x = xp[n*3+0], y = xp[n*3+1], z = xp[n*3+2];
            float w = mp[n];
            float wx = w * x, wy = w * y;
            sxx += wx*x; sxy += wx*y; sxz += wx*z;
            syy += wy*y; syz += wy*z; szz += w*z*z;
        }
        // I = tr*I3 - outer  (tr = sxx+syy+szz)
        float a00 = syy + szz, a11 = sxx + szz, a22 = sxx + syy;
        float a01 = -sxy, a02 = -sxz, a12 = -syz;

        // eigenvalues: trigonometric method (ascending like jnp.linalg.eigh)
        float q   = (a00 + a11 + a22) * (1.0f/3.0f);
        float b00 = a00 - q, b11 = a11 - q, b22 = a22 - q;
        float p1 = a01*a01 + a02*a02 + a12*a12;
        float p2 = b00*b00 + b11*b11 + b22*b22 + 2.f*p1;
        float p  = sqrtf(p2 * (1.0f/6.0f)) + 1e-30f;
        float ip = 1.0f / p;
        float c00 = b00*ip, c01 = a01*ip, c02 = a02*ip;
        float c11 = b11*ip, c12 = a12*ip, c22 = b22*ip;
        float detB = c00*(c11*c22 - c12*c12)
                   - c01*(c01*c22 - c12*c02)
                   + c02*(c01*c12 - c11*c02);
        float rr  = fminf(1.0f, fmaxf(-1.0f, 0.5f * detB));
        float phi = acosf(rr) * (1.0f/3.0f);
        float emax = q + 2.f*p*cosf(phi);
        float emin = q + 2.f*p*cosf(phi + 2.0943951023931953f); // + 2*pi/3
        float emid = 3.f*q - emax - emin;

        float v1x,v1y,v1z, v2x,v2y,v2z;
        eigvec3(a00,a01,a02,a11,a12,a22, emin, v1x,v1y,v1z);
        eigvec3(a00,a01,a02,a11,a12,a22, emid, v2x,v2y,v2z);
        float d12 = v1x*v2x + v1y*v2y + v1z*v2z;       // Gram-Schmidt
        v2x -= d12*v1x; v2y -= d12*v1y; v2z -= d12*v1z;
        float inv2 = rsqrtf(v2x*v2x + v2y*v2y + v2z*v2z + 1e-30f);
        v2x *= inv2; v2y *= inv2; v2z *= inv2;
        // col 2 = v1 x v2  => det(rot) = +1 (reference's inversion fix)
        float v3x = v1y*v2z - v1z*v2y;
        float v3y = v1z*v2x - v1x*v2z;
        float v3z = v1x*v2y - v1y*v2x;

        float* rp = &s_rot[(bl*4 + r) * 9];   // rot[i][j], columns = eigenvectors
        rp[0] = v1x; rp[1] = v2x; rp[2] = v3x;
        rp[3] = v1y; rp[4] = v2y; rp[5] = v3y;
        rp[6] = v1z; rp[7] = v2z; rp[8] = v3z;
    }
    __syncthreads();

    // ---- phase 3: out = X * rot for all 4 rotations via V_WMMA_F32_16X16X4_F32 ----
    // A (16x4): rows = 16 points, K = (x,y,z,0).  B (4x16): col c=3r+j holds rot[r][:,j].
    {
        int wave   = tid >> 5;
        int lane   = tid & 31;
        int hi     = lane >> 4;         // 0: K={0,1}  1: K={2,3}
        int mrow   = lane & 15;         // A row / D column index
        int c      = mrow;              // B/D column
        bool cval  = c < 12;
        int cc     = cval ? c : 11;     // clamp for safe LDS addressing
        int r      = cc / 3;
        int j      = cc - r * 3;
        // loop-invariant pieces
        int  a_off   = hi * 2;          // x or z within point
        int  b_off   = hi * 6;          // rot row 0 or row 2
        int  rj      = r * 9 + j;       // s_rot offset within batch
        int  rbase   = r * 96 + j;      // s_out offset within batch
        int  mrow3   = mrow * 3;
        float fhi    = hi ? 0.0f : 1.0f;    // kills K=1 slot in hi half
        float fval   = cval ? 1.0f : 0.0f;  // kills padding columns 12..15
        #pragma unroll
        for (int k = 0; k < 8; ++k) {
            int t     = wave * 8 + k;   // 64 tiles per block
            int bl    = t >> 1;
            int half  = (t & 1) * 16;
            // A operand: unconditional loads, computed index (no exec-masked loads)
            const float* xp = &s_xyz[bl * XS + half * 3 + mrow3];
            float ax = xp[a_off];       // V0: K=0 | K=2
            float ay = xp[1];           // K=1 value
            v2f a;
            a.x = ax;
            a.y = ay * fhi;             // V1: K=1 | K=3(pad)=0
            // B operand
            const float* rp = &s_rot[bl * 36 + rj];
            float blo = rp[b_off];      // rot[0][j] | rot[2][j]
            float bmid = rp[3];         // rot[1][j]
            v2f bv;
            bv.x = blo * fval;
            bv.y = bmid * fhi * fval;
            v8f acc = {0.f,0.f,0.f,0.f,0.f,0.f,0.f,0.f};
            v8f d = __builtin_amdgcn_wmma_f32_16x16x4_f32(
                        false, a, false, bv, (short)0, acc, false, false);
            if (cval) {
                int base = bl * 384 + rbase;
                int nb   = half + hi * 8;            // D: VGPR v -> row v | v+8
                #pragma unroll
                for (int v = 0; v < 8; ++v)
                    s_out[base + (nb + v) * 3] = d[v];
            }
        }
    }
    __syncthreads();

    // ---- phase 4: fully coalesced b128 store of the contiguous (32,4,32,3) tile ----
    float4* go = (float4*)(out + b0 * 384);
    #pragma unroll
    for (int it = 0; it < 12; ++it) {
        int i = tid + it * 256;          // 3072 float4s
        go[i] = *(const float4*)&s_out[i * 4];
    }
}

extern "C" void kernel_launch(void* const* d_in, const int* in_sizes, int n_in,
                              void* d_out, int out_size, void* d_ws, size_t ws_size,
                              hipStream_t stream) {
    const float* xyz  = (const float*)d_in[0];   // (B, 32, 3) f32
    const float* mass = (const float*)d_in[1];   // (4, 32)    f32
    float* out = (float*)d_out;                  // (B, 4, 32, 3) f32
    int B = in_sizes[0] / 96;                    // 200000
    int nblocks = B / BB;                        // 6250 (B divisible by 32)
    inertia_rot_kernel<<<nblocks, 256, 0, stream>>>(xyz, mass, out);
}